// PPI_72129680769752
// MI455X (gfx1250) — compile-verified
//
#include <hip/hip_runtime.h>
#include <math.h>

// ---- problem constants (match reference) ----
#define N_NODES 40000
#define N_EDGES 640000
#define ETOT    (N_EDGES + N_NODES)   // with self loops
#define F_IN    50
#define H1      4
#define OC1     64
#define CH      256                    // H1*OC1
#define H3      6
#define C_OUT   121
#define SCAN_B  512                    // block size for CSR prefix scan

typedef __attribute__((ext_vector_type(2)))  float    v2f;
typedef __attribute__((ext_vector_type(8)))  float    v8f;

__device__ inline void edge_sd(const int* __restrict__ ei, int e,
                               int& s, int& d) {
    if (e < N_EDGES) { s = ei[e]; d = ei[N_EDGES + e]; }
    else             { s = d = e - N_EDGES; }
}

// -------------------------------------------------------------------------
// weight transpose: W[K x Nc] -> Wt[Nc x K] (column-major B for the GEMM)
// -------------------------------------------------------------------------
__global__ void transpose_w(const float* __restrict__ W, float* __restrict__ Wt,
                            int K, int Nc)
{
    int i = blockIdx.x * blockDim.x + threadIdx.x;
    if (i >= K * Nc) return;
    int k = i / Nc, n = i - k * Nc;
    Wt[(size_t)n * K + k] = W[i];
}

// -------------------------------------------------------------------------
// WMMA GEMM: C[M x Nc] = A[M x K] @ B[K x Nc], fp32.
// A row-major, Bt column-major (Nc x K) so both operand fragments are
// contiguous b64 loads with immediate offsets. One wave computes a 16x64 C
// tile. K is processed in chunks of 16: a load phase (20 b64 loads into
// explicitly live fragment arrays -> one s_clause) followed by a compute
// phase (16 WMMAs), so loadcnt waits retire in staggered steps instead of
// a full drain per matrix op. KMAIN (48 / 256) is a multiple of 16.
// -------------------------------------------------------------------------
template <int K>
__global__ void __launch_bounds__(128)
gemm_wmma_f32(const float* __restrict__ A, const float* __restrict__ Bt,
              float* __restrict__ C, int M, int Nc)
{
    const int lane = threadIdx.x & 31;
    const int wave = threadIdx.x >> 5;          // 0..3
    const int half = lane >> 4;                 // 0/1: lanes 0-15 | 16-31
    const int l16  = lane & 15;

    const int row0 = (blockIdx.y * 4 + wave) * 16;
    const int col0 = blockIdx.x * 64;
    if (row0 >= M) return;                      // wave-uniform exit

    // fragment base pointers (half*2 baked in: lanes 16-31 hold K+2,K+3)
    const float* Ap = A + (size_t)(row0 + l16) * K + half * 2;
    int  bc[4]; bool bv[4]; const float* Bp[4];
    #pragma unroll
    for (int j = 0; j < 4; ++j) {
        bc[j] = col0 + 16 * j + l16;
        bv[j] = (bc[j] < Nc);
        Bp[j] = Bt + (size_t)(bv[j] ? bc[j] : 0) * K + half * 2;  // clamped
    }

    v8f acc[4];
    #pragma unroll
    for (int j = 0; j < 4; ++j) { v8f z = {}; acc[j] = z; }

    constexpr int KMAIN = K & ~3;               // 48 or 256
    static_assert(KMAIN % 16 == 0, "chunking assumes KMAIN % 16 == 0");

    for (int kc = 0; kc < KMAIN; kc += 16) {
        // ---- load phase: 4 K-steps of fragments, all live at once ----
        v2f af[4], bf[4][4];
        #pragma unroll
        for (int t = 0; t < 4; ++t) {
            af[t] = *(const v2f*)(Ap + kc + 4 * t);
            #pragma unroll
            for (int j = 0; j < 4; ++j)
                bf[t][j] = *(const v2f*)(Bp[j] + kc + 4 * t);
        }
        // ---- compute phase: 16 WMMAs ----
        #pragma unroll
        for (int t = 0; t < 4; ++t) {
            #pragma unroll
            for (int j = 0; j < 4; ++j)
                acc[j] = __builtin_amdgcn_wmma_f32_16x16x4_f32(
                             false, af[t], false, bf[t][j], (short)0,
                             acc[j], false, false);
        }
    }

    if constexpr (KMAIN < K) {
        // K tail: clamped-address loads + VALU selects (branch-free zero pad)
        const float* ArowRaw = A + (size_t)(row0 + l16) * K;
        const int ka0 = KMAIN + half * 2, ka1 = ka0 + 1;
        const int ca0 = ka0 < K ? ka0 : K - 1;
        const int ca1 = ka1 < K ? ka1 : K - 1;
        v2f at;
        float a0 = ArowRaw[ca0], a1 = ArowRaw[ca1];
        at.x = (ka0 < K) ? a0 : 0.0f;
        at.y = (ka1 < K) ? a1 : 0.0f;
        #pragma unroll
        for (int j = 0; j < 4; ++j) {
            const float* BcolRaw = Bt + (size_t)(bv[j] ? bc[j] : 0) * K;
            float p0 = BcolRaw[ca0], p1 = BcolRaw[ca1];
            v2f btl;
            btl.x = (ka0 < K) ? p0 : 0.0f;
            btl.y = (ka1 < K) ? p1 : 0.0f;
            acc[j] = __builtin_amdgcn_wmma_f32_16x16x4_f32(
                         false, at, false, btl, (short)0, acc[j], false, false);
        }
    }

    // C/D layout: VGPR r -> rows (r, r+8) for lane halves (0,1)
    #pragma unroll
    for (int r = 0; r < 8; ++r) {
        const size_t row = (size_t)(row0 + r + half * 8) * Nc;
        #pragma unroll
        for (int j = 0; j < 4; ++j)
            if (bv[j]) C[row + bc[j]] = acc[j][r];
    }
}

// -------------------------------------------------------------------------
// CSR-by-destination build (graph is shared by all three layers)
// -------------------------------------------------------------------------
__global__ void fill_u32(int* __restrict__ p, int v, int cnt) {
    int i = blockIdx.x * blockDim.x + threadIdx.x;
    if (i < cnt) p[i] = v;
}

__global__ void csr_count(const int* __restrict__ ei, int* __restrict__ cnt) {
    int e = blockIdx.x * blockDim.x + threadIdx.x;
    if (e >= ETOT) return;
    int s, d; edge_sd(ei, e, s, d);
    atomicAdd(&cnt[d], 1);
}

// block-level exclusive scan: excl[i] = local exclusive prefix, bsum[b] = block total
__global__ void __launch_bounds__(SCAN_B)
csr_scan_block(const int* __restrict__ cnt, int* __restrict__ excl,
               int* __restrict__ bsum)
{
    __shared__ int sh[SCAN_B];
    const int g = blockIdx.x * SCAN_B + threadIdx.x;
    const int v = (g < N_NODES) ? cnt[g] : 0;
    sh[threadIdx.x] = v;
    __syncthreads();
    for (int off = 1; off < SCAN_B; off <<= 1) {
        int t = (threadIdx.x >= off) ? sh[threadIdx.x - off] : 0;
        __syncthreads();
        sh[threadIdx.x] += t;
        __syncthreads();
    }
    const int incl = sh[threadIdx.x];
    if (g < N_NODES) excl[g] = incl - v;
    if (threadIdx.x == SCAN_B - 1) bsum[blockIdx.x] = incl;
}

__global__ void csr_scan_top(int* __restrict__ bsum, int nb) {
    if (blockIdx.x == 0 && threadIdx.x == 0) {
        int run = 0;
        for (int i = 0; i < nb; ++i) { int v = bsum[i]; bsum[i] = run; run += v; }
    }
}

__global__ void csr_finalize(const int* __restrict__ excl,
                             const int* __restrict__ bsum,
                             int* __restrict__ indptr, int* __restrict__ cur)
{
    int i = blockIdx.x * blockDim.x + threadIdx.x;
    if (i > N_NODES) return;
    if (i == N_NODES) { indptr[N_NODES] = ETOT; return; }
    int v = excl[i] + bsum[i / SCAN_B];
    indptr[i] = v;
    cur[i] = v;
}

__global__ void csr_scatter(const int* __restrict__ ei, int* __restrict__ cur,
                            int* __restrict__ srcs)
{
    int e = blockIdx.x * blockDim.x + threadIdx.x;
    if (e >= ETOT) return;
    int s, d; edge_sd(ei, e, s, d);
    int pos = atomicAdd(&cur[d], 1);
    srcs[pos] = s;
}

// -------------------------------------------------------------------------
// attention
// -------------------------------------------------------------------------
__global__ void attn_coeff(const float* __restrict__ HF,
                           const float* __restrict__ as_,
                           const float* __restrict__ ad_,
                           float* __restrict__ AL, float* __restrict__ AR,
                           int Hh, int OCc)
{
    int i = blockIdx.x * blockDim.x + threadIdx.x;   // i = node*Hh + h
    if (i >= N_NODES * Hh) return;
    int h = i % Hh;
    const float* hp = HF + (size_t)i * OCc;
    float sl = 0.0f, sr = 0.0f;
    for (int c = 0; c < OCc; ++c) {
        float v = hp[c];
        sl += v * as_[h * OCc + c];
        sr += v * ad_[h * OCc + c];
    }
    AL[i] = sl; AR[i] = sr;
}

__device__ inline float lrelu02(float x) { return x > 0.0f ? x : 0.2f * x; }

// One wave per (dst, head): segment softmax via shuffle reductions, then
// register-accumulated weighted feature sum. Zero atomics, coalesced store.
template <int OCc, int Hh>
__global__ void __launch_bounds__(256)
gat_aggregate_csr(const int* __restrict__ indptr, const int* __restrict__ srcs,
                  const float* __restrict__ AL, const float* __restrict__ AR,
                  const float* __restrict__ HF, float* __restrict__ AGG)
{
    constexpr int NCH = (OCc + 31) / 32;
    const int wv   = (int)((blockIdx.x * blockDim.x + threadIdx.x) >> 5);
    const int lane = threadIdx.x & 31;
    if (wv >= N_NODES * Hh) return;              // wave-uniform
    const int dst = wv / Hh, h = wv - dst * Hh;
    const int beg = indptr[dst], end = indptr[dst + 1];  // >=1 edge (self loop)
    const float ar = AR[wv];

    // pass 1: segment max (lanes over edges)
    float m = -3.0e38f;
    for (int i = beg + lane; i < end; i += 32)
        m = fmaxf(m, lrelu02(AL[srcs[i] * Hh + h] + ar));
    #pragma unroll
    for (int off = 16; off; off >>= 1) m = fmaxf(m, __shfl_xor(m, off, 32));

    // pass 2: segment sum of exp
    float ssum = 0.0f;
    for (int i = beg + lane; i < end; i += 32)
        ssum += expf(lrelu02(AL[srcs[i] * Hh + h] + ar) - m);
    #pragma unroll
    for (int off = 16; off; off >>= 1) ssum += __shfl_xor(ssum, off, 32);
    const float inv = 1.0f / ssum;

    // pass 3: weighted feature accumulation (lanes over channels)
    float acc[NCH];
    #pragma unroll
    for (int j = 0; j < NCH; ++j) acc[j] = 0.0f;
    for (int i = beg; i < end; ++i) {
        const int s = srcs[i];                                   // wave-uniform
        const float alpha = expf(lrelu02(AL[s * Hh + h] + ar) - m) * inv;
        const float* sp = HF + ((size_t)s * Hh + h) * OCc;
        #pragma unroll
        for (int j = 0; j < NCH; ++j) {
            const int c = lane + 32 * j;
            if (c < OCc) acc[j] += alpha * sp[c];
        }
    }
    float* dp = AGG + ((size_t)dst * Hh + h) * OCc;
    #pragma unroll
    for (int j = 0; j < NCH; ++j) {
        const int c = lane + 32 * j;
        if (c < OCc) dp[c] = acc[j];
    }
}

// -------------------------------------------------------------------------
// epilogues
// -------------------------------------------------------------------------
__global__ void finalize_concat(const float* __restrict__ AGG,
                                const float* __restrict__ bg,
                                const float* __restrict__ LIN,
                                const float* __restrict__ bl,
                                float* __restrict__ out, int total, int CHc)
{
    int i = blockIdx.x * blockDim.x + threadIdx.x;
    if (i >= total) return;
    int c = i % CHc;
    float v = AGG[i] + bg[c] + LIN[i] + bl[c];
    out[i] = v > 0.0f ? v : (expf(v) - 1.0f);
}

__global__ void finalize_mean(const float* __restrict__ AGG,
                              const float* __restrict__ bg,
                              const float* __restrict__ LIN,
                              const float* __restrict__ bl,
                              float* __restrict__ out)
{
    int i = blockIdx.x * blockDim.x + threadIdx.x;
    if (i >= N_NODES * C_OUT) return;
    int node = i / C_OUT, c = i - node * C_OUT;
    float s = 0.0f;
    #pragma unroll
    for (int h = 0; h < H3; ++h)
        s += AGG[((size_t)node * H3 + h) * C_OUT + c];
    out[i] = s * (1.0f / (float)H3) + bg[c] + LIN[i] + bl[c];
}

// -------------------------------------------------------------------------
// host-side driver
// -------------------------------------------------------------------------
static void launch_gemm(hipStream_t st, const float* A, const float* Bt,
                        float* C, int M, int K, int Nc)
{
    dim3 grid((Nc + 63) / 64, ((M + 15) / 16 + 3) / 4);
    if (K == F_IN)
        gemm_wmma_f32<F_IN><<<grid, 128, 0, st>>>(A, Bt, C, M, Nc);
    else
        gemm_wmma_f32<CH><<<grid, 128, 0, st>>>(A, Bt, C, M, Nc);
}

static void launch_transpose(hipStream_t st, const float* W, float* Wt,
                             int K, int Nc)
{
    int n = K * Nc;
    transpose_w<<<(n + 255) / 256, 256, 0, st>>>(W, Wt, K, Nc);
}

extern "C" void kernel_launch(void* const* d_in, const int* in_sizes, int n_in,
                              void* d_out, int out_size, void* d_ws, size_t ws_size,
                              hipStream_t stream)
{
    const float* x   = (const float*)d_in[0];
    const int*   ei  = (const int*)  d_in[1];
    const float* W1  = (const float*)d_in[2];
    const float* a1s = (const float*)d_in[3];
    const float* a1d = (const float*)d_in[4];
    const float* b1  = (const float*)d_in[5];
    const float* Wl1 = (const float*)d_in[6];
    const float* bl1 = (const float*)d_in[7];
    const float* W2  = (const float*)d_in[8];
    const float* a2s = (const float*)d_in[9];
    const float* a2d = (const float*)d_in[10];
    const float* b2  = (const float*)d_in[11];
    const float* Wl2 = (const float*)d_in[12];
    const float* bl2 = (const float*)d_in[13];
    const float* W3  = (const float*)d_in[14];
    const float* a3s = (const float*)d_in[15];
    const float* a3d = (const float*)d_in[16];
    const float* b3  = (const float*)d_in[17];
    const float* Wl3 = (const float*)d_in[18];
    const float* bl3 = (const float*)d_in[19];

    float* out    = (float*)d_out;                       // [N, 121]
    float* hidden = out + (size_t)N_NODES * C_OUT;       // [N, 256]

    // ---- workspace layout ----
    float* p    = (float*)d_ws;
    float* HF   = p; p += (size_t)N_NODES * H3 * C_OUT;  // N*726 head features
    float* AGG  = p; p += (size_t)N_NODES * H3 * C_OUT;  // N*726 aggregation
    float* LIN  = p; p += (size_t)N_NODES * CH;          // skip-connection GEMM out
    float* FEAT = p; p += (size_t)N_NODES * CH;          // layer-1 output features
    float* AL   = p; p += (size_t)N_NODES * H3;
    float* AR   = p; p += (size_t)N_NODES * H3;
    // transposed weights (column-major B operands)
    float* W1t  = p; p += CH * F_IN;
    float* Wl1t = p; p += CH * F_IN;
    float* W2t  = p; p += CH * CH;
    float* Wl2t = p; p += CH * CH;
    float* W3t  = p; p += (H3 * C_OUT) * CH;
    float* Wl3t = p; p += C_OUT * CH;
    int* ip     = (int*)p;
    int* CNT    = ip; ip += N_NODES;
    int* EXCL   = ip; ip += N_NODES;
    int* INDPTR = ip; ip += N_NODES + 1;
    int* CUR    = ip; ip += N_NODES;
    int* BSUM   = ip; ip += 256;                         // >= #scan blocks (79)
    int* SRCS   = ip; ip += ETOT;

    const int nch   = N_NODES * CH;
    const int nscan = (N_NODES + SCAN_B - 1) / SCAN_B;

    // ---- transpose weights (tiny; done every call, graph-capture safe) ----
    launch_transpose(stream, W1,  W1t,  F_IN, CH);
    launch_transpose(stream, Wl1, Wl1t, F_IN, CH);
    launch_transpose(stream, W2,  W2t,  CH, CH);
    launch_transpose(stream, Wl2, Wl2t, CH, CH);
    launch_transpose(stream, W3,  W3t,  CH, H3 * C_OUT);
    launch_transpose(stream, Wl3, Wl3t, CH, C_OUT);

    // ---- build CSR by destination (shared by all layers) ----
    fill_u32<<<(N_NODES + 255) / 256, 256, 0, stream>>>(CNT, 0, N_NODES);
    csr_count<<<(ETOT + 255) / 256, 256, 0, stream>>>(ei, CNT);
    csr_scan_block<<<nscan, SCAN_B, 0, stream>>>(CNT, EXCL, BSUM);
    csr_scan_top<<<1, 32, 0, stream>>>(BSUM, nscan);
    csr_finalize<<<(N_NODES + 256) / 256, 256, 0, stream>>>(EXCL, BSUM, INDPTR, CUR);
    csr_scatter<<<(ETOT + 255) / 256, 256, 0, stream>>>(ei, CUR, SRCS);

    // ---------------- layer 1: F_IN -> 4 heads x 64, concat, ELU -----------
    launch_gemm(stream, x, W1t,  HF,  N_NODES, F_IN, CH);
    launch_gemm(stream, x, Wl1t, LIN, N_NODES, F_IN, CH);
    attn_coeff<<<(N_NODES * H1 + 255) / 256, 256, 0, stream>>>(HF, a1s, a1d, AL, AR, H1, OC1);
    gat_aggregate_csr<OC1, H1><<<(N_NODES * H1 + 7) / 8, 256, 0, stream>>>(
        INDPTR, SRCS, AL, AR, HF, AGG);
    finalize_concat<<<(nch + 255) / 256, 256, 0, stream>>>(
        AGG, b1, LIN, bl1, FEAT, nch, CH);

    // ---------------- layer 2: CH -> 4 heads x 64, concat, ELU -> hidden ---
    launch_gemm(stream, FEAT, W2t,  HF,  N_NODES, CH, CH);
    launch_gemm(stream, FEAT, Wl2t, LIN, N_NODES, CH, CH);
    attn_coeff<<<(N_NODES * H1 + 255) / 256, 256, 0, stream>>>(HF, a2s, a2d, AL, AR, H1, OC1);
    gat_aggregate_csr<OC1, H1><<<(N_NODES * H1 + 7) / 8, 256, 0, stream>>>(
        INDPTR, SRCS, AL, AR, HF, AGG);
    finalize_concat<<<(nch + 255) / 256, 256, 0, stream>>>(
        AGG, b2, LIN, bl2, hidden, nch, CH);

    // ---------------- layer 3: CH -> 6 heads x 121, mean, no act -> out ----
    launch_gemm(stream, hidden, W3t,  HF,  N_NODES, CH, H3 * C_OUT);
    launch_gemm(stream, hidden, Wl3t, LIN, N_NODES, CH, C_OUT);
    attn_coeff<<<(N_NODES * H3 + 255) / 256, 256, 0, stream>>>(HF, a3s, a3d, AL, AR, H3, C_OUT);
    gat_aggregate_csr<C_OUT, H3><<<(N_NODES * H3 + 7) / 8, 256, 0, stream>>>(
        INDPTR, SRCS, AL, AR, HF, AGG);
    finalize_mean<<<(N_NODES * C_OUT + 255) / 256, 256, 0, stream>>>(
        AGG, b3, LIN, bl3, out);
}